// TreeMultiQuad_55190329753602
// MI455X (gfx1250) — compile-verified
//
#include <hip/hip_runtime.h>

#define SZ   4096
#define HALF 2048
#define NPIX (SZ * SZ)

typedef int   v4i __attribute__((ext_vector_type(4)));
typedef float v4f __attribute__((ext_vector_type(4)));

// ---- analytic inverse of jnp.nonzero enumeration (row-major) ----

// filled-site rank j -> flat pixel index, per channel (0=R,1=G,2=B)
template <int CH>
__device__ __forceinline__ int decode_filled(int j) {
    int jr = j >> 11;        // j / 2048
    int ji = j & 2047;       // j % 2048
    if (CH == 0) {           // R: (even row, even col)
        return (jr << 13) + (ji << 1);
    } else if (CH == 2) {    // B: (odd row, odd col)
        return (jr << 13) + SZ + (ji << 1) + 1;
    } else {                 // G: even row -> odd col, odd row -> even col
        int c = (jr & 1) ? (ji << 1) : ((ji << 1) + 1);
        return (jr << 12) + c;
    }
}

// unfilled-site rank u -> flat pixel index, per channel
template <int CH>
__device__ __forceinline__ int decode_unfilled(unsigned u) {
    if (CH == 1) {           // G unfilled: even row -> even col, odd row -> odd col
        int r = (int)(u >> 11);
        int i = (int)(u & 2047u);
        int c = (r & 1) ? ((i << 1) + 1) : (i << 1);
        return (r << 12) + c;
    }
    // R/B: 6144 unfilled sites per row-pair
    unsigned p2  = u / 6144u;
    unsigned rem = u - p2 * 6144u;
    int rowbase = (int)(p2 << 13);          // (2*p2) * 4096
    if (CH == 0) {                          // R
        if (rem < HALF) return rowbase + (int)((rem << 1) + 1);      // even row, odd cols
        else            return rowbase + SZ + (int)(rem - HALF);     // odd row, all cols
    } else {                                // B
        if (rem < SZ)   return rowbase + (int)rem;                   // even row, all cols
        else            return rowbase + SZ + (int)((rem - SZ) << 1);// odd row, even cols
    }
}

// ---- copy raw values to each pixel's own channel (every pixel is a filled
//      site of exactly one channel in the Bayer pattern) ----
__global__ void fill_kernel(const float* __restrict__ img, float* __restrict__ out) {
    int t  = blockIdx.x * blockDim.x + threadIdx.x;   // one thread per 4 pixels
    int p0 = t << 2;
    v4f v  = *((const v4f*)img + t);                  // regular (temporal) load: seed L2 with image
    int rodd = (p0 >> 12) & 1;
    float vals[4] = {v.x, v.y, v.z, v.w};
#pragma unroll
    for (int k = 0; k < 4; ++k) {
        int p    = p0 + k;
        int codd = p & 1;
        int ch   = rodd ? (codd ? 2 : 1) : (codd ? 1 : 0);
        // NT store: output is write-once, keep L2 for the image plane
        __builtin_nontemporal_store(vals[k], out + (size_t)ch * NPIX + p);
    }
}

// ---- weighted gather-fill for one channel's unfilled sites ----
template <int CH>
__global__ void interp_kernel(const float* __restrict__ img,
                              const int*   __restrict__ idx,
                              const float* __restrict__ dist,
                              float*       __restrict__ out,
                              int n) {
    int stride = gridDim.x * blockDim.x;
    for (int u = blockIdx.x * blockDim.x + threadIdx.x; u < n; u += stride) {
        int un = u + stride;
        if (un < n) {
            // gfx1250 global_prefetch_b8: pull next iteration's streaming lines
            __builtin_prefetch((const void*)(idx  + 4 * (size_t)un), 0, 0);
            __builtin_prefetch((const void*)(dist + 4 * (size_t)un), 0, 0);
        }
        // 128-bit NT streaming loads (don't evict the 64MB image from 192MB L2)
        v4i id = __builtin_nontemporal_load((const v4i*)idx  + u);
        v4f w  = __builtin_nontemporal_load((const v4f*)dist + u);

        // gathers: random within the 64MB image plane -> L2 hits
        float v0 = img[decode_filled<CH>(id.x)];
        float v1 = img[decode_filled<CH>(id.y)];
        float v2 = img[decode_filled<CH>(id.z)];
        float v3 = img[decode_filled<CH>(id.w)];

        float num = v0 * w.x + v1 * w.y + v2 * w.z + v3 * w.w;
        float den = w.x + w.y + w.z + w.w;
        float val = num / den;

        int p = decode_unfilled<CH>((unsigned)u);
        __builtin_nontemporal_store(val, out + (size_t)CH * NPIX + p);
    }
}

extern "C" void kernel_launch(void* const* d_in, const int* in_sizes, int n_in,
                              void* d_out, int out_size, void* d_ws, size_t ws_size,
                              hipStream_t stream) {
    const float* img    = (const float*)d_in[0];   // coded [1,1,4096,4096]
    const int*   idx_r  = (const int*)  d_in[1];
    const float* dist_r = (const float*)d_in[2];
    const int*   idx_g  = (const int*)  d_in[3];
    const float* dist_g = (const float*)d_in[4];
    const int*   idx_b  = (const int*)  d_in[5];
    const float* dist_b = (const float*)d_in[6];
    float*       out    = (float*)d_out;           // [3, 4096, 4096]

    int n_u_rb = in_sizes[1] / 4;   // 12,582,912 unfilled R (and B) sites
    int n_u_g  = in_sizes[3] / 4;   //  8,388,608 unfilled G sites

    // 1) seed output with raw values at each channel's own sites (also warms L2 with img)
    fill_kernel<<<NPIX / 4 / 256, 256, 0, stream>>>(img, out);

    // 2) gather-interpolate the holes per channel (disjoint writes, any order)
    interp_kernel<0><<<4096, 256, 0, stream>>>(img, idx_r, dist_r, out, n_u_rb);
    interp_kernel<1><<<4096, 256, 0, stream>>>(img, idx_g, dist_g, out, n_u_g);
    interp_kernel<2><<<4096, 256, 0, stream>>>(img, idx_b, dist_b, out, n_u_rb);
}